// UnifiedGaussianAdapter_43508018709010
// MI455X (gfx1250) — compile-verified
//
#include <hip/hip_runtime.h>
#include <hip/hip_bf16.h>
#include <stdint.h>

#define TILE 256
#define DIN 55                       // 7 + 3*16
#define TILE_FLOATS (TILE * DIN)     // 14080 floats = 56320 bytes
#define COV_OFF (0)                  // staging: cov  = TILE*9 floats
#define SCL_OFF (TILE * 9)           // staging: scal = TILE*3 floats (total 3072 < 14080)

#ifndef __has_builtin
#define __has_builtin(x) 0
#endif

#if __has_builtin(__builtin_amdgcn_tensor_load_to_lds) && __has_builtin(__builtin_amdgcn_s_wait_tensorcnt)
#define USE_TDM 1
#else
#define USE_TDM 0
#endif

typedef __attribute__((ext_vector_type(4))) unsigned int u32x4;
typedef __attribute__((ext_vector_type(8))) int          i32x8;
typedef __attribute__((ext_vector_type(4))) int          i32x4;

// SH degree mask: deg0=1, deg1(1..3)=0.1*0.25, deg2(4..8)=0.1*0.25^2, deg3(9..15)=0.1*0.25^3
__constant__ float c_sh_mask[16] = {
    1.0f,
    0.025f, 0.025f, 0.025f,
    0.00625f, 0.00625f, 0.00625f, 0.00625f, 0.00625f,
    0.0015625f, 0.0015625f, 0.0015625f, 0.0015625f, 0.0015625f, 0.0015625f, 0.0015625f
};

// Per-gaussian math. Stores opacity (stride-1) and rotation (float4) directly
// (both naturally coalesced); returns cov[9] and scales[3] in registers.
__device__ __forceinline__ void compute_one(const float* __restrict__ r, long long g,
                                            float* __restrict__ o_opac,
                                            float* __restrict__ o_rot,
                                            float cov[9], float sc[3])
{
    const float eps = 1e-8f;

    // opacity = sigmoid(raw[0])
    o_opac[g] = 1.0f / (1.0f + __expf(-r[0]));

    // scales = min(0.001 * softplus(raw[1:4]), 0.3), stable softplus
#pragma unroll
    for (int k = 0; k < 3; ++k) {
        float x = r[1 + k];
        float sp = fmaxf(x, 0.0f) + log1pf(__expf(-fabsf(x)));
        sc[k] = fminf(0.001f * sp, 0.3f);
    }

    // rotations output = RAW quaternion (unnormalized)
    float qi = r[4], qj = r[5], qk = r[6], qr = r[7];
    float4 q4; q4.x = qi; q4.y = qj; q4.z = qk; q4.w = qr;
    ((float4*)o_rot)[g] = q4;

    // normalize: q / (||q|| + eps)
    float n2  = qi * qi + qj * qj + qk * qk + qr * qr;
    float inv = 1.0f / (sqrtf(n2) + eps);
    float i_ = qi * inv, j_ = qj * inv, k_ = qk * inv, r_ = qr * inv;

    // quaternion_to_matrix: two_s = 2 / (dot(qn,qn) + eps)
    float d = i_ * i_ + j_ * j_ + k_ * k_ + r_ * r_;
    float ts = 2.0f / (d + eps);

    float R00 = 1.0f - ts * (j_ * j_ + k_ * k_);
    float R01 = ts * (i_ * j_ - k_ * r_);
    float R02 = ts * (i_ * k_ + j_ * r_);
    float R10 = ts * (i_ * j_ + k_ * r_);
    float R11 = 1.0f - ts * (i_ * i_ + k_ * k_);
    float R12 = ts * (j_ * k_ - i_ * r_);
    float R20 = ts * (i_ * k_ - j_ * r_);
    float R21 = ts * (j_ * k_ + i_ * r_);
    float R22 = 1.0f - ts * (i_ * i_ + j_ * j_);

    // RS = R * diag(scales); cov = RS * RS^T (symmetric)
    float a0 = R00 * sc[0], a1 = R01 * sc[1], a2 = R02 * sc[2];
    float b0 = R10 * sc[0], b1 = R11 * sc[1], b2 = R12 * sc[2];
    float c0 = R20 * sc[0], c1 = R21 * sc[1], c2 = R22 * sc[2];

    float c00 = a0 * a0 + a1 * a1 + a2 * a2;
    float c01 = a0 * b0 + a1 * b1 + a2 * b2;
    float c02 = a0 * c0 + a1 * c1 + a2 * c2;
    float c11 = b0 * b0 + b1 * b1 + b2 * b2;
    float c12 = b0 * c0 + b1 * c1 + b2 * c2;
    float c22 = c0 * c0 + c1 * c1 + c2 * c2;
    cov[0] = c00; cov[1] = c01; cov[2] = c02;
    cov[3] = c01; cov[4] = c11; cov[5] = c12;
    cov[6] = c02; cov[7] = c12; cov[8] = c22;
}

__global__ __launch_bounds__(TILE)
void gaussian_adapter_kernel(const float* __restrict__ raw,
                             float* __restrict__ out,
                             long long G, int numTiles)
{
    __shared__ __align__(16) float lds[TILE_FLOATS];
    const int tid = threadIdx.x;

    float* o_cov  = out + 3  * G;
    float* o_sh   = out + 12 * G;
    float* o_opac = out + 60 * G;
    float* o_scal = out + 61 * G;
    float* o_rot  = out + 64 * G;

    for (int tile = blockIdx.x; tile < numTiles; tile += gridDim.x) {
        const long long base = (long long)tile * TILE;
        const long long rem  = G - base;
        const int cnt = (rem < (long long)TILE) ? (int)rem : TILE;

        // Warm GL2 with the *next* tile this block will process.
        const int ntile = tile + gridDim.x;
        if (ntile < numTiles) {
            const char* np = (const char*)(raw + (long long)ntile * TILE * DIN);
            for (int off = tid * 128; off < TILE_FLOATS * 4; off += TILE * 128)
                __builtin_prefetch(np + off, 0, 1);
        }

        if (cnt == TILE) {
#if USE_TDM
            // One TDM descriptor DMA per tile: 14080 contiguous f32 -> LDS.
            // Issued by wave 0 only (TDM ignores EXEC; branch is wave-uniform).
            if ((tid >> 5) == 0) {
                unsigned long long ga = (unsigned long long)(uintptr_t)(const void*)(raw + base * DIN);
                unsigned ldsoff = (unsigned)(uintptr_t)(void*)&lds[0];   // low 32 bits = LDS offset
                // D# group0: count=1(valid), lds_addr, global_addr[56:0], type=2
                u32x4 g0 = { 1u,
                             ldsoff,
                             (unsigned)(ga & 0xFFFFFFFFull),
                             (unsigned)((ga >> 32) & 0x1FFFFFFull) | (2u << 30) };
                // D# group1: data_size=4B (code 2); 1-D tensor_dim0 = tile_dim0 = 14080;
                // tensor_dim0_stride = 14080; mask/pad/iterate/barrier = 0.
                const unsigned td0 = (unsigned)TILE_FLOATS;
                i32x8 g1 = { (int)(2u << 16),                 // w0: data_size
                             (int)((td0 & 0xFFFFu) << 16),    // w1: tensor_dim0[15:0] @ bits63:48
                             (int)((td0 >> 16) & 0xFFFFu),    // w2: tensor_dim0[31:16]
                             (int)((td0 & 0xFFFFu) << 16),    // w3: tile_dim0 @ bits127:112
                             0,                               // w4: tile_dim1/2 unused
                             (int)td0,                        // w5: tensor_dim0_stride[31:0]
                             0, 0 };                          // w6,w7
                i32x4 gz4 = { 0, 0, 0, 0 };                   // groups 2/3: dims 2..4 unused
                i32x8 gz8 = { 0, 0, 0, 0, 0, 0, 0, 0 };      // trailing group (6-arg toolchain form)
                __builtin_amdgcn_tensor_load_to_lds(g0, g1, gz4, gz4, gz8, 0);
                __builtin_amdgcn_s_wait_tensorcnt(0);
            }
#else
            // Cooperative coalesced float4 staging fallback.
            const float4* src = (const float4*)(raw + base * DIN);
            float4* dst = (float4*)lds;
            for (int i = tid; i < TILE_FLOATS / 4; i += TILE) dst[i] = src[i];
#endif
            __syncthreads();   // (1) tile data visible to all waves

            // Per-thread gaussian: tid*55 LDS addressing is bank-conflict-free (55 coprime 64).
            float cov[9], sc[3];
            compute_one(lds + tid * DIN, base + tid, o_opac, o_rot, cov, sc);

            // SH = scaled copy of raw[8:56]; fully coalesced float4 stores
            // (48 % 4 == 0 so each float4 stays inside one gaussian).
            float4* shdst = (float4*)(o_sh + base * 48);
            for (int i = tid; i < (TILE * 48) / 4; i += TILE) {
                const int f  = i * 4;
                const int gl = f / 48;
                const int c  = f - gl * 48;
                const float* rb = lds + gl * DIN + 8 + c;
                float4 v;
                v.x = rb[0] * c_sh_mask[(c)     & 15];
                v.y = rb[1] * c_sh_mask[(c + 1) & 15];
                v.z = rb[2] * c_sh_mask[(c + 2) & 15];
                v.w = rb[3] * c_sh_mask[(c + 3) & 15];
                shdst[i] = v;
            }
            __syncthreads();   // (2) all reads of the raw tile complete

            // Transpose cov/scales through LDS so every global store is coalesced.
            // Strides 9 and 3 are coprime with 64 banks -> conflict-free ds writes.
#pragma unroll
            for (int k = 0; k < 9; ++k) lds[COV_OFF + tid * 9 + k] = cov[k];
#pragma unroll
            for (int k = 0; k < 3; ++k) lds[SCL_OFF + tid * 3 + k] = sc[k];
            __syncthreads();   // (3) staging visible

            float4* covdst = (float4*)(o_cov + base * 9);
            const float4* covsrc = (const float4*)(lds + COV_OFF);
            for (int i = tid; i < (TILE * 9) / 4; i += TILE) covdst[i] = covsrc[i];

            float4* scldst = (float4*)(o_scal + base * 3);
            const float4* sclsrc = (const float4*)(lds + SCL_OFF);
            for (int i = tid; i < (TILE * 3) / 4; i += TILE) scldst[i] = sclsrc[i];

            __syncthreads();   // (4) LDS reads done before next tile's DMA overwrites
        } else {
            // Ragged tail (not hit for G = 2^21, kept for generality): direct path.
            if (tid < cnt) {
                const long long g = base + tid;
                const float* r = raw + g * DIN;
                float cov[9], sc[3];
                compute_one(r, g, o_opac, o_rot, cov, sc);
                for (int k = 0; k < 9; ++k) o_cov[g * 9 + k] = cov[k];
                for (int k = 0; k < 3; ++k) o_scal[g * 3 + k] = sc[k];
                for (int c = 0; c < 48; ++c)
                    o_sh[g * 48 + c] = r[8 + c] * c_sh_mask[c & 15];
            }
        }
    }
}

extern "C" void kernel_launch(void* const* d_in, const int* in_sizes, int n_in,
                              void* d_out, int out_size, void* d_ws, size_t ws_size,
                              hipStream_t stream)
{
    const float* means = (const float*)d_in[0];
    const float* raw   = (const float*)d_in[1];
    float* out = (float*)d_out;

    const long long G = (long long)in_sizes[0] / 3;

    // means output is an identity pass-through: peak-BW D2D copy on the stream.
    (void)hipMemcpyAsync(out, means, (size_t)in_sizes[0] * sizeof(float),
                         hipMemcpyDeviceToDevice, stream);

    const int numTiles = (int)((G + TILE - 1) / TILE);
    int blocks = numTiles < 4096 ? numTiles : 4096;
    if (blocks < 1) blocks = 1;

    gaussian_adapter_kernel<<<dim3(blocks), dim3(TILE), 0, stream>>>(raw, out, G, numTiles);
}